// instance_attention_layer_41171556499651
// MI455X (gfx1250) — compile-verified
//
#include <hip/hip_runtime.h>
#include <hip/hip_bf16.h>

typedef __attribute__((ext_vector_type(2))) float v2f;
typedef __attribute__((ext_vector_type(8))) float v8f;

#define N_CLASS 32
#define N_SUP   16
#define FDIM    1024
#define N_QRY   512
#define QTILE   16
#define DCHUNK  256
#define LSTRIDE (DCHUNK + 4)   // 260 % 64 banks = 4 -> 16 rows hit distinct bank groups

__device__ __forceinline__ float dev_tanh(float x) {
#if __has_builtin(__builtin_amdgcn_tanhf)
  return __builtin_amdgcn_tanhf(x);        // v_tanh_f32 on gfx1250
#else
  return tanhf(x);
#endif
}

__global__ __launch_bounds__(256)
void proto_att_kernel(const float* __restrict__ data, float* __restrict__ out) {
  __shared__ float sQ[QTILE * LSTRIDE];
  __shared__ float sS[N_SUP * LSTRIDE];
  __shared__ float sAtt[QTILE][N_SUP + 1];

  const int qt  = blockIdx.x;   // query tile 0..31
  const int c   = blockIdx.y;   // class 0..31
  const int tid = threadIdx.x;  // 0..255

  const float* Qg = data + (size_t)(N_CLASS * N_SUP + qt * QTILE) * FDIM;
  const float* Sg = data + (size_t)c * N_SUP * FDIM;

  const int q   = tid >> 4;     // 0..15
  const int s   = tid & 15;     // 0..15
  const int row = tid >> 4;     // cooperative-load row
  const int l16 = tid & 15;

  // ---- Phase 1: scores[q][s] = sum_d tanh(S[s][d] * Q[q][d]), D streamed in chunks
  float acc = 0.f;
  for (int d0 = 0; d0 < FDIM; d0 += DCHUNK) {
    const float4* qsrc = (const float4*)(Qg + (size_t)row * FDIM + d0);
    const float4* ssrc = (const float4*)(Sg + (size_t)row * FDIM + d0);
    float4* qdst = (float4*)(sQ + row * LSTRIDE);
    float4* sdst = (float4*)(sS + row * LSTRIDE);
#pragma unroll
    for (int j = 0; j < 4; ++j) {
      int c4 = l16 + j * 16;     // float4 index within 256-float row chunk
      qdst[c4] = qsrc[c4];
      sdst[c4] = ssrc[c4];
    }
    __syncthreads();
    const float* qr = sQ + q * LSTRIDE;   // broadcast within half-wave
    const float* sr = sS + s * LSTRIDE;   // 16 rows -> distinct banks
#pragma unroll 8
    for (int d = 0; d < DCHUNK; d += 4) {
      float4 qv = *(const float4*)(qr + d);
      float4 sv = *(const float4*)(sr + d);
      acc += dev_tanh(qv.x * sv.x);
      acc += dev_tanh(qv.y * sv.y);
      acc += dev_tanh(qv.z * sv.z);
      acc += dev_tanh(qv.w * sv.w);
    }
    __syncthreads();
  }
  sAtt[q][s] = acc;
  __syncthreads();

  // ---- Phase 2: softmax over s (each thread redundantly reduces its 16-wide row)
  float m = -3.402823466e38f;
  float rowv[N_SUP];
#pragma unroll
  for (int i = 0; i < N_SUP; ++i) { rowv[i] = sAtt[q][i]; m = fmaxf(m, rowv[i]); }
  float sum = 0.f;
#pragma unroll
  for (int i = 0; i < N_SUP; ++i) sum += __expf(rowv[i] - m);
  float mya = __expf(acc - m) / sum;
  __syncthreads();
  sAtt[q][s] = mya;
  __syncthreads();

  // ---- Phase 3: proto tile = att(16x16) @ support_c(16x1024) via V_WMMA_F32_16X16X4_F32
  const int wave = tid >> 5;    // 0..7
  const int lane = tid & 31;
  const int hi   = lane >> 4;   // lane group selects K pair per ISA f32 layouts
  const int ln   = lane & 15;

  // A fragments: 16x4 f32 per K-step; VGPR0 = {K0 | K2}, VGPR1 = {K1 | K3}
  v2f a[4];
#pragma unroll
  for (int kk = 0; kk < 4; ++kk) {
    a[kk].x = sAtt[ln][kk * 4 + (hi ? 2 : 0)];
    a[kk].y = sAtt[ln][kk * 4 + (hi ? 3 : 1)];
  }

  float* outBase = out + ((size_t)(qt * QTILE) * N_CLASS + c) * FDIM;

  for (int nt = wave; nt < FDIM / 16; nt += 8) {   // wave-uniform: EXEC stays all-ones
    const int n0 = nt * 16;
    v8f dacc = {};
#pragma unroll
    for (int kk = 0; kk < 4; ++kk) {
      // B fragment 4x16: lanes = N, VGPR0 = {K0 | K2}, VGPR1 = {K1 | K3}; L2-resident
      v2f b;
      b.x = Sg[(size_t)(kk * 4 + (hi ? 2 : 0)) * FDIM + n0 + ln];
      b.y = Sg[(size_t)(kk * 4 + (hi ? 3 : 1)) * FDIM + n0 + ln];
      dacc = __builtin_amdgcn_wmma_f32_16x16x4_f32(false, a[kk], false, b,
                                                   (short)0, dacc, false, false);
    }
    // C/D layout: VGPR r -> M = r (+8 for hi half), N = lane%16
#pragma unroll
    for (int r = 0; r < 8; ++r) {
      int M = r + (hi ? 8 : 0);
      outBase[(size_t)M * N_CLASS * FDIM + n0 + ln] = dacc[r];
    }
  }
}

extern "C" void kernel_launch(void* const* d_in, const int* in_sizes, int n_in,
                              void* d_out, int out_size, void* d_ws, size_t ws_size,
                              hipStream_t stream) {
  const float* data = (const float*)d_in[0];
  float* out = (float*)d_out;
  (void)in_sizes; (void)n_in; (void)d_ws; (void)ws_size; (void)out_size;
  dim3 grid(N_QRY / QTILE, N_CLASS);
  proto_att_kernel<<<grid, 256, 0, stream>>>(data, out);
}